// DarcyLoss_54614804136400
// MI455X (gfx1250) — compile-verified
//
#include <hip/hip_runtime.h>
#include <hip/hip_bf16.h>
#include <stdint.h>

// DarcyLoss for MI455X (gfx1250).
// Memory-bound: 201 MB minimum HBM traffic -> ~8.6 us at 23.3 TB/s.
// CDNA5 paths: async global->LDS tile staging (ASYNCcnt) for the stencil
// kernel, and V_WMMA_F32_16X16X4_F32 as a cross-lane reduction engine.

namespace {
constexpr int Hn = 512, Wn = 512, Bn = 32;
constexpr int TILE = 32, HALO = 2;
constexpr int LREG = TILE + 2 * HALO;   // 36 = loaded region per tile
constexpr int LP   = 40;                // padded LDS pitch (avoid bank conflicts)
constexpr int TPB  = 256;               // 8 waves (wave32)
constexpr int NTX  = Wn / TILE;         // 16
constexpr int NTY  = Hn / TILE;         // 16
constexpr int NB_MSE   = 2048;
constexpr int NB_DARCY = Bn * NTX * NTY; // 8192
constexpr int NPART    = NB_MSE + NB_DARCY;
}

typedef float v2f __attribute__((ext_vector_type(2)));
typedef float v8f __attribute__((ext_vector_type(8)));

// ---- CDNA5 async global->LDS copy (per-lane, tracked by ASYNCcnt) ----------
// GVS addressing: saddr = 64-bit SGPR plane base, vaddr = 32-bit byte offset,
// vdst = per-lane LDS byte address.
__device__ __forceinline__ void async_ld_b32(uint32_t lds_off, uint32_t goff,
                                             const float* base) {
  asm volatile("global_load_async_to_lds_b32 %0, %1, %2"
               :: "v"(lds_off), "v"(goff),
                  "s"((unsigned long long)(uintptr_t)base)
               : "memory");
}

__device__ __forceinline__ void wait_async_all() {
  asm volatile("s_wait_asynccnt 0" ::: "memory");
}

// Low 32 bits of a generic pointer to __shared__ == wave-relative LDS byte
// address (LDS aperture is 4GB-aligned; flat->LDS mapping truncates to [31:0]).
__device__ __forceinline__ uint32_t lds_addr_of(const void* p) {
  return (uint32_t)(uintptr_t)p;
}

// ---- WMMA-based block reduction --------------------------------------------
// Per wave: A = ones(16x4), B = 4x16 holding the 32 lane partials (other half
// zero). D(0,n) = sum_k B(k,n), so sum over row M=0 (c[0], lanes 0..15) equals
// the wave total regardless of the exact B lane mapping. One v_wmma collapses
// 32 lane partials -> 16 values; a short LDS tree finishes across 8 waves.
// `red` must hold >= 128 floats. EXEC must be all ones (uniform control flow).
__device__ __forceinline__ float block_reduce_wmma(float acc, float* red) {
  v2f a; a[0] = 1.0f; a[1] = 1.0f;
  v2f b; b[0] = acc;  b[1] = 0.0f;
  v8f c = {};
  c = __builtin_amdgcn_wmma_f32_16x16x4_f32(false, a, false, b,
                                            (short)0, c, false, false);
  const int t = threadIdx.x;
  const int lane = t & 31, wid = t >> 5;
  __syncthreads();                 // prior uses of `red` memory complete
  if (lane < 16) red[wid * 16 + lane] = c[0];
  __syncthreads();
  #pragma unroll
  for (int s = 64; s > 0; s >>= 1) {   // reduce 8 waves * 16 = 128 values
    if (t < s) red[t] += red[t + s];
    __syncthreads();
  }
  return red[0];
}

// ---- edge_order=2 gradient stencils (exact match to torch.gradient) --------
__device__ __forceinline__ float d_dx(const float* s, int c, int gx) {
  if (gx == 0)      return -1.5f * s[c] + 2.0f * s[c + 1] - 0.5f * s[c + 2];
  if (gx == Wn - 1) return  1.5f * s[c] - 2.0f * s[c - 1] + 0.5f * s[c - 2];
  return 0.5f * (s[c + 1] - s[c - 1]);
}
__device__ __forceinline__ float d_dy(const float* s, int c, int gy) {
  if (gy == 0)      return -1.5f * s[c] + 2.0f * s[c + LP] - 0.5f * s[c + 2 * LP];
  if (gy == Hn - 1) return  1.5f * s[c] - 2.0f * s[c - LP] + 0.5f * s[c - 2 * LP];
  return 0.5f * (s[c + LP] - s[c - LP]);
}
__device__ __forceinline__ float ux_at(const float* sK, const float* sP, int c, int gx) {
  return -sK[c] * d_dx(sP, c, gx);
}
__device__ __forceinline__ float uy_at(const float* sK, const float* sP, int c, int gy) {
  return -sK[c] * d_dy(sP, c, gy);
}

// ---- Kernel: Darcy residual F^2 partials (async LDS tile staging) ----------
__global__ void __launch_bounds__(TPB)
darcy_kernel(const float* __restrict__ x0, const float* __restrict__ sigma,
             float* __restrict__ ws) {
  __shared__ float sK[LREG * LP];
  __shared__ float sP[LREG * LP];

  const int t = threadIdx.x;
  const int tileX = blockIdx.x, tileY = blockIdx.y, b = blockIdx.z;
  const float* Kg = x0 + (size_t)b * 2 * Hn * Wn;   // channel 0
  const float* Pg = Kg + (size_t)Hn * Wn;           // channel 1
  const int gy0 = tileY * TILE - HALO;
  const int gx0 = tileX * TILE - HALO;
  const uint32_t baseK = lds_addr_of(&sK[0]);
  const uint32_t baseP = lds_addr_of(&sP[0]);

  // Stage 36x36 K and p tiles into LDS with CDNA5 async loads. Halo cells
  // outside the plane are clamped; the edge-formula selection below never
  // reads them, so the values are don't-care.
  for (int idx = t; idx < LREG * LREG; idx += TPB) {
    int ry = idx / LREG;
    int rx = idx - ry * LREG;
    int gy = gy0 + ry; gy = gy < 0 ? 0 : (gy > Hn - 1 ? Hn - 1 : gy);
    int gx = gx0 + rx; gx = gx < 0 ? 0 : (gx > Wn - 1 ? Wn - 1 : gx);
    uint32_t goff = (uint32_t)((gy * Wn + gx) * 4);
    uint32_t loff = (uint32_t)((ry * LP + rx) * 4);
    async_ld_b32(baseK + loff, goff, Kg);
    async_ld_b32(baseP + loff, goff, Pg);
  }
  wait_async_all();   // drain this wave's ASYNCcnt
  __syncthreads();    // make all waves' LDS writes visible

  // Each thread computes 4 output points of the 32x32 tile.
  const int lx = t & (TILE - 1);
  const int ly_base = (t >> 5) * 4;
  const int gx = tileX * TILE + lx;
  float acc = 0.0f;
  #pragma unroll
  for (int i = 0; i < 4; ++i) {
    const int ly = ly_base + i;
    const int gy = tileY * TILE + ly;
    const int c = (ly + HALO) * LP + (lx + HALO);
    float dx, dy;
    if (gx == 0)
      dx = -1.5f * ux_at(sK, sP, c, 0) + 2.0f * ux_at(sK, sP, c + 1, 1)
           - 0.5f * ux_at(sK, sP, c + 2, 2);
    else if (gx == Wn - 1)
      dx =  1.5f * ux_at(sK, sP, c, Wn - 1) - 2.0f * ux_at(sK, sP, c - 1, Wn - 2)
           + 0.5f * ux_at(sK, sP, c - 2, Wn - 3);
    else
      dx = 0.5f * (ux_at(sK, sP, c + 1, gx + 1) - ux_at(sK, sP, c - 1, gx - 1));

    if (gy == 0)
      dy = -1.5f * uy_at(sK, sP, c, 0) + 2.0f * uy_at(sK, sP, c + LP, 1)
           - 0.5f * uy_at(sK, sP, c + 2 * LP, 2);
    else if (gy == Hn - 1)
      dy =  1.5f * uy_at(sK, sP, c, Hn - 1) - 2.0f * uy_at(sK, sP, c - LP, Hn - 2)
           + 0.5f * uy_at(sK, sP, c - 2 * LP, Hn - 3);
    else
      dy = 0.5f * (uy_at(sK, sP, c + LP, gy + 1) - uy_at(sK, sP, c - LP, gy - 1));

    float src = 0.0f;
    if (gy < 64 && gx < 64) src = 10.0f;                     // w = round(0.125*512)
    else if (gy >= Hn - 64 && gx >= Wn - 64) src = -10.0f;

    float F = -(dx + dy) + src;
    acc += F * F;
  }

  // WMMA cross-lane reduction (reuses sK; helper barriers before overwrite).
  float total = block_reduce_wmma(acc, sK);
  if (t == 0) {
    // contribution weight: 0.5 * (sigma/0.01) / (H*W) / B  =  50*sigma/(H*W*B)
    float scale = 50.0f * sigma[b] * (1.0f / ((float)Hn * (float)Wn * (float)Bn));
    ws[NB_MSE + ((b * NTY + tileY) * NTX + tileX)] = total * scale;
  }
}

// ---- Kernel: MSE partials --------------------------------------------------
__global__ void __launch_bounds__(TPB)
mse_kernel(const float* __restrict__ a, const float* __restrict__ b,
           float* __restrict__ ws) {
  const float4* A = (const float4*)a;
  const float4* Bv = (const float4*)b;
  constexpr int n4 = (Bn * 2 * Hn * Wn) / 4;  // 4,194,304 -> exactly 8 iters/thread
  float acc = 0.0f;
  for (int i = blockIdx.x * TPB + threadIdx.x; i < n4; i += NB_MSE * TPB) {
    float4 x = A[i], y = Bv[i];
    float d0 = x.x - y.x, d1 = x.y - y.y, d2 = x.z - y.z, d3 = x.w - y.w;
    acc += d0 * d0 + d1 * d1 + d2 * d2 + d3 * d3;
  }
  __shared__ float red[128];
  float total = block_reduce_wmma(acc, red);
  if (threadIdx.x == 0)
    ws[blockIdx.x] = total * (1.0f / (float)(Bn * 2 * Hn * Wn));
}

// ---- Kernel: final reduction of all partials -> scalar ---------------------
__global__ void __launch_bounds__(TPB)
final_kernel(const float* __restrict__ ws, float* __restrict__ out) {
  float acc = 0.0f;
  for (int i = threadIdx.x; i < NPART; i += TPB) acc += ws[i];
  __shared__ float red[128];
  float total = block_reduce_wmma(acc, red);
  if (threadIdx.x == 0) out[0] = total;
}

extern "C" void kernel_launch(void* const* d_in, const int* in_sizes, int n_in,
                              void* d_out, int out_size, void* d_ws, size_t ws_size,
                              hipStream_t stream) {
  const float* model_output = (const float*)d_in[0];
  const float* target       = (const float*)d_in[1];
  const float* x0_hat       = (const float*)d_in[2];
  const float* sigma_t      = (const float*)d_in[3];
  float* out = (float*)d_out;
  float* ws  = (float*)d_ws;   // needs NPART (=10240) floats = 40 KB scratch

  mse_kernel<<<NB_MSE, TPB, 0, stream>>>(model_output, target, ws);
  darcy_kernel<<<dim3(NTX, NTY, Bn), TPB, 0, stream>>>(x0_hat, sigma_t, ws);
  final_kernel<<<1, TPB, 0, stream>>>(ws, out);
}